// QNNClassifier_25735444037790
// MI455X (gfx1250) — compile-verified
//
#include <hip/hip_runtime.h>
#include <hip/hip_bf16.h>

typedef __attribute__((ext_vector_type(2))) float v2f;
typedef __attribute__((ext_vector_type(8))) float v8f;

// ---------------------------------------------------------------------------
// Prep kernel: build the batch-shared 4x4 complex circuit unitary U from the
// 12 quantum params (2 layers x 2 wires x (phi,theta,omega)), exactly as
// _circuit_unitary does. One thread; ~300 flops total. Writes Re(U) rows to
// Umat[0..15], Im(U) rows to Umat[16..31] (row-major) in d_ws.
// ---------------------------------------------------------------------------
__device__ inline void rotmat2(float phi, float th, float om,
                               float Rre[2][2], float Rim[2][2]) {
    float c = cosf(0.5f * th), s = sinf(0.5f * th);
    float ap = 0.5f * (phi + om);   // ep = e^{-i*ap}
    float am = 0.5f * (phi - om);   // em = e^{+i*am}
    float ep_re = cosf(ap), ep_im = -sinf(ap);
    float em_re = cosf(am), em_im =  sinf(am);
    Rre[0][0] =  ep_re * c;  Rim[0][0] =  ep_im * c;   // ep*c
    Rre[0][1] = -em_re * s;  Rim[0][1] = -em_im * s;   // -em*s
    Rre[1][0] =  em_re * s;  Rim[1][0] = -em_im * s;   // conj(em)*s
    Rre[1][1] =  ep_re * c;  Rim[1][1] = -ep_im * c;   // conj(ep)*c
}

__global__ void qnn_prep(const float* __restrict__ qp, float* __restrict__ Umat) {
    if (threadIdx.x != 0 || blockIdx.x != 0) return;
    float Ure[4][4] = {}, Uim[4][4] = {};
    for (int i = 0; i < 4; ++i) Ure[i][i] = 1.f;

    for (int l = 0; l < 2; ++l) {
        float r0re[2][2], r0im[2][2], r1re[2][2], r1im[2][2];
        rotmat2(qp[l*6 + 0], qp[l*6 + 1], qp[l*6 + 2], r0re, r0im);
        rotmat2(qp[l*6 + 3], qp[l*6 + 4], qp[l*6 + 5], r1re, r1im);

        // K = kron(r0, r1) (complex)
        float Kre[4][4], Kim[4][4];
        for (int i = 0; i < 2; ++i)
            for (int k = 0; k < 2; ++k)
                for (int j = 0; j < 2; ++j)
                    for (int m = 0; m < 2; ++m) {
                        Kre[2*i+k][2*j+m] = r0re[i][j]*r1re[k][m] - r0im[i][j]*r1im[k][m];
                        Kim[2*i+k][2*j+m] = r0re[i][j]*r1im[k][m] + r0im[i][j]*r1re[k][m];
                    }
        // M = K @ U (complex 4x4)
        float Mre[4][4], Mim[4][4];
        for (int i = 0; i < 4; ++i)
            for (int j = 0; j < 4; ++j) {
                float ar = 0.f, ai = 0.f;
                for (int k = 0; k < 4; ++k) {
                    ar += Kre[i][k]*Ure[k][j] - Kim[i][k]*Uim[k][j];
                    ai += Kre[i][k]*Uim[k][j] + Kim[i][k]*Ure[k][j];
                }
                Mre[i][j] = ar; Mim[i][j] = ai;
            }
        // CNOT01 @ M : row permutation {0,1,3,2}; then CNOT10 @ : {0,3,2,1}
        const int p1[4] = {0, 1, 3, 2};
        const int p2[4] = {0, 3, 2, 1};
        float Tre[4][4], Tim[4][4];
        for (int i = 0; i < 4; ++i)
            for (int j = 0; j < 4; ++j) { Tre[i][j] = Mre[p1[i]][j]; Tim[i][j] = Mim[p1[i]][j]; }
        for (int i = 0; i < 4; ++i)
            for (int j = 0; j < 4; ++j) { Ure[i][j] = Tre[p2[i]][j]; Uim[i][j] = Tim[p2[i]][j]; }
    }
    for (int i = 0; i < 4; ++i)
        for (int j = 0; j < 4; ++j) {
            Umat[i*4 + j]      = Ure[i][j];
            Umat[16 + i*4 + j] = Uim[i][j];
        }
}

// ---------------------------------------------------------------------------
// Main streaming kernel. Per 16 batch elements, one V_WMMA_F32_16X16X4_F32:
//   A (16x4):  rows 0-3 = Re(U), rows 4-7 = Im(U), rows 8-15 = 0 (loop-invariant)
//   B (4x16):  column n = psi0 of element n (lane n gives k=0,1; lane n+16 k=2,3)
//   D (16x16): lane n (<16) holds re0..re3,im0..im3 of element n in its 8 accum
//              VGPRs -> ev and the Linear(1,1) are lane-local, store coalesced.
// Memory bound: 12 B/element -> the WMMA cost is noise; it replaces ~30 VALU
// FMAs per element with the matrix pipe, overlapping the HBM stream.
// ---------------------------------------------------------------------------
__global__ __launch_bounds__(256) void qnn_main(const float* __restrict__ x,
                                                const float* __restrict__ Umat,
                                                const float* __restrict__ fc_w,
                                                const float* __restrict__ fc_b,
                                                float* __restrict__ out,
                                                int nTiles) {
    const int lane = threadIdx.x & 31;
    const int hi   = lane >> 4;     // 0: K=0,1 half | 1: K=2,3 half
    const int rA   = lane & 15;     // A row / batch sub-index within tile

    // Loop-invariant A operand: [Re(U); Im(U); 0] rows.
    v2f a;
    if (rA < 8) {
        a.x = Umat[rA*4 + hi*2 + 0];
        a.y = Umat[rA*4 + hi*2 + 1];
    } else {
        a.x = 0.f; a.y = 0.f;
    }
    const float w = fc_w[0];
    const float b = fc_b[0];

    const int waveId = (int)((blockIdx.x * blockDim.x + threadIdx.x) >> 5);
    const int nWaves = (int)((gridDim.x * blockDim.x) >> 5);

    for (int t = waveId; t < nTiles; t += nWaves) {
        const int e = t * 16 + rA;
        // lanes n and n+16 read the same float2 -> one 128B line per tile
        float2 xv = ((const float2*)x)[e];
        // prefetch next tile for this wave (global_prefetch_b8, speculative)
        __builtin_prefetch(&((const float2*)x)[e + nWaves * 16], 0, 0);

        float s0, c0, s1, c1;
        __sincosf(0.5f * xv.x, &s0, &c0);
        __sincosf(0.5f * xv.y, &s1, &c1);
        const float p0 = c0 * c1, p1 = c0 * s1, p2 = s0 * c1, p3 = s0 * s1;

        v2f bm;
        bm.x = hi ? p2 : p0;
        bm.y = hi ? p3 : p1;

        v8f d = {};
        d = __builtin_amdgcn_wmma_f32_16x16x4_f32(
                /*neg_a=*/false, a, /*neg_b=*/false, bm,
                /*c_mod=*/(short)0, d, /*reuse_a=*/false, /*reuse_b=*/false);

        // lane<16: d[k]=Re(psi_k), d[k+4]=Im(psi_k) of element e
        float ev = (d[0]*d[0] + d[4]*d[4]) + (d[1]*d[1] + d[5]*d[5])
                 - (d[2]*d[2] + d[6]*d[6]) - (d[3]*d[3] + d[7]*d[7]);
        if (lane < 16) out[e] = ev * w + b;
    }
}

// ---------------------------------------------------------------------------
// Launch: inputs are (x[B,2] f32, q_params[2,2,3] f32, fc_w[1,1] f32, fc_b[1] f32).
// Output: [B,1] f32. U (32 floats) staged in d_ws.
// ---------------------------------------------------------------------------
extern "C" void kernel_launch(void* const* d_in, const int* in_sizes, int n_in,
                              void* d_out, int out_size, void* d_ws, size_t ws_size,
                              hipStream_t stream) {
    const float* x    = (const float*)d_in[0];
    const float* qp   = (const float*)d_in[1];
    const float* fc_w = (const float*)d_in[2];
    const float* fc_b = (const float*)d_in[3];
    float* out  = (float*)d_out;
    float* Umat = (float*)d_ws;            // 32 floats

    const int nbatch = in_sizes[0] / 2;    // 2,097,152
    const int nTiles = (nbatch + 15) / 16; // 131,072

    qnn_prep<<<1, 32, 0, stream>>>(qp, Umat);

    const int threads = 256;               // 8 wave32 per block
    const int blocks  = 1024;              // 8192 waves, 16 tiles/wave
    qnn_main<<<blocks, threads, 0, stream>>>(x, Umat, fc_w, fc_b, out, nTiles);
}